// GNNModel_19774029431469
// MI455X (gfx1250) — compile-verified
//
#include <hip/hip_runtime.h>

typedef float v2f_t __attribute__((ext_vector_type(2)));
typedef float v8f_t __attribute__((ext_vector_type(8)));

static __device__ __forceinline__ void atomic_fadd(float* p, float v) {
  __hip_atomic_fetch_add(p, v, __ATOMIC_RELAXED, __HIP_MEMORY_SCOPE_AGENT);
}

__global__ void zero_f32(float* __restrict__ p, long long n) {
  long long t = (long long)blockIdx.x * blockDim.x + threadIdx.x;
  if (t < n) p[t] = 0.0f;
}

// deg[dst[e]] += 1
__global__ void deg_accum(const int* __restrict__ dst, float* __restrict__ deg, long long E) {
  long long t = (long long)blockIdx.x * blockDim.x + threadIdx.x;
  if (t < E) atomic_fadd(&deg[dst[t]], 1.0f);
}

// d[i] = rsqrt(deg[i] + 1)  (self-loop included in degree)
__global__ void deg_to_dinv(float* __restrict__ d, long long n) {
  long long t = (long long)blockIdx.x * blockDim.x + threadIdx.x;
  if (t < n) d[t] = rsqrtf(d[t] + 1.0f);
}

// H[n x NCOL] = X[n x K] @ W[K x NCOL], via V_WMMA_F32_16X16X4_F32.
// One wave computes a 16-row x 16-col tile. K padded to mult of 4 and N padded
// to 16 at COMPILE TIME; out-of-range elements are fetched from a clamped safe
// address and zeroed with a select (no EXEC divergence -> unconditional loads).
// Store epilogue: wave-uniform fast path for full tiles (stores with immediate
// offsets off one base pointer), guarded path only for the single tail tile.
template <int K, int NCOL>
__global__ void gemm_wmma_f32(const float* __restrict__ X, const float* __restrict__ W,
                              float* __restrict__ H, int n) {
  constexpr int KT = (K + 3) / 4;
  int lane = threadIdx.x & 31;
  int wave = (int)((blockIdx.x * (long long)blockDim.x + threadIdx.x) >> 5);
  int row_base = wave * 16;
  if (row_base >= n) return;          // wave-uniform: EXEC stays all-1s for WMMA

  int half = lane >> 4;               // lane group 0..15 / 16..31
  int l    = lane & 15;

  int m = row_base + l;
  if (m > n - 1) m = n - 1;           // clamp tail reads (rows never stored)
  const float* __restrict__ xrow = X + (long long)m * K;

  const bool colok = (NCOL == 16) ? true : (l < NCOL);  // folds for NCOL==16
  const int  lc    = colok ? l : 0;

  v8f_t c = {};
#pragma unroll
  for (int kt = 0; kt < KT; ++kt) {
    int k0 = kt * 4 + 2 * half;       // this lane half supplies rows k0, k0+1
    bool ok0 = (k0     < K);
    bool ok1 = (k0 + 1 < K);
    int ka0 = ok0 ? k0     : 0;       // clamped safe addresses
    int ka1 = ok1 ? k0 + 1 : 0;

    float a0 = xrow[ka0];
    float a1 = xrow[ka1];
    float w0 = W[ka0 * NCOL + lc];
    float w1 = W[ka1 * NCOL + lc];

    v2f_t a, b;
    a.x = ok0 ? a0 : 0.0f;
    a.y = ok1 ? a1 : 0.0f;
    b.x = (ok0 && colok) ? w0 : 0.0f;
    b.y = (ok1 && colok) ? w1 : 0.0f;

    // D = A(16x4) * B(4x16) + C
    c = __builtin_amdgcn_wmma_f32_16x16x4_f32(false, a, false, b, (short)0, c, false, false);
  }

  // C/D layout: VGPR v -> M = v + 8*half, N = l
  float* __restrict__ hb = H + (long long)(row_base + 8 * half) * NCOL + l;
  if (row_base + 16 <= n) {           // wave-uniform fast path: full tile
    if (colok) {
#pragma unroll
      for (int v = 0; v < 8; ++v) hb[v * NCOL] = c[v];
    }
  } else {                            // tail tile only
#pragma unroll
    for (int v = 0; v < 8; ++v) {
      int row = row_base + 8 * half + v;
      if (row < n && colok) hb[v * NCOL] = c[v];
    }
  }
}

// agg[dst,f] += h[src,f] * dinv[src]*dinv[dst]; one thread per (edge, feature)
__global__ void scatter_edges(const float* __restrict__ H, const float* __restrict__ dinv,
                              const int* __restrict__ src, const int* __restrict__ dst,
                              float* __restrict__ agg, long long E, int logF) {
  long long t = (long long)blockIdx.x * blockDim.x + threadIdx.x;
  if (t >= (E << logF)) return;
  long long e = t >> logF;
  int f = (int)(t & ((1 << logF) - 1));
  int s = src[e], d = dst[e];
  float cf = dinv[s] * dinv[d];
  atomic_fadd(&agg[((long long)d << logF) + f], H[((long long)s << logF) + f] * cf);
}

// agg = relu(agg + h*dinv^2 + b)   (in place)
__global__ void finalize_relu(const float* __restrict__ H, const float* __restrict__ dinv,
                              const float* __restrict__ b, float* __restrict__ agg,
                              long long n, int logF) {
  long long t = (long long)blockIdx.x * blockDim.x + threadIdx.x;
  if (t >= (n << logF)) return;
  long long i = t >> logF;
  int f = (int)(t & ((1 << logF) - 1));
  float dv = dinv[i];
  float v = agg[t] + H[t] * dv * dv + b[f];
  agg[t] = v > 0.0f ? v : 0.0f;
}

// out[i] = relu(agg2 + h2*dinv^2 + b2) . Wd + bd
__global__ void head_kernel(const float* __restrict__ H2, const float* __restrict__ agg2,
                            const float* __restrict__ dinv, const float* __restrict__ b2,
                            const float* __restrict__ Wd, const float* __restrict__ bd,
                            float* __restrict__ out, long long n) {
  long long i = (long long)blockIdx.x * blockDim.x + threadIdx.x;
  if (i >= n) return;
  float dv = dinv[i];
  float d2 = dv * dv;
  float acc = bd[0];
#pragma unroll
  for (int f = 0; f < 8; ++f) {
    float v = agg2[i * 8 + f] + H2[i * 8 + f] * d2 + b2[f];
    v = v > 0.0f ? v : 0.0f;
    acc += v * Wd[f];
  }
  out[i] = acc;
}

extern "C" void kernel_launch(void* const* d_in, const int* in_sizes, int n_in,
                              void* d_out, int out_size, void* d_ws, size_t ws_size,
                              hipStream_t stream) {
  const float* x  = (const float*)d_in[0];
  const int*   ei = (const int*)d_in[1];
  const float* W1 = (const float*)d_in[2];
  const float* b1 = (const float*)d_in[3];
  const float* W2 = (const float*)d_in[4];
  const float* b2 = (const float*)d_in[5];
  const float* Wd = (const float*)d_in[6];
  const float* bd = (const float*)d_in[7];
  float* out = (float*)d_out;

  long long n = in_sizes[0] / 10;
  long long E = in_sizes[1] / 2;
  const int* src = ei;
  const int* dst = ei + E;

  // Workspace layout (floats): [dinv : n] [bufH : n*16] [bufA : n*16]
  // Layer 2 reuses bufH: h2 = bufH[0..n*8), agg2 = bufH[n*8..n*16)
  float* ws = (float*)d_ws;
  long long o_h = ((n + 63) / 64) * 64;
  float* dinv = ws;
  float* h1   = ws + o_h;
  float* agg1 = ws + o_h + n * 16;
  float* h2   = h1;
  float* agg2 = h1 + n * 8;

  const int B = 256;
  auto nb = [](long long t, int b) { return (unsigned)((t + b - 1) / b); };

  // --- degrees -> dinv ---
  zero_f32<<<nb(n, B), B, 0, stream>>>(dinv, n);
  deg_accum<<<nb(E, B), B, 0, stream>>>(dst, dinv, E);
  deg_to_dinv<<<nb(n, B), B, 0, stream>>>(dinv, n);

  long long waves = (n + 15) / 16;          // one wave per 16-row tile
  long long gthreads = waves * 32;

  // --- layer 1: h1 = x @ W1 (K=10, N=16) ---
  gemm_wmma_f32<10, 16><<<nb(gthreads, B), B, 0, stream>>>(x, W1, h1, (int)n);
  zero_f32<<<nb(n * 16, B), B, 0, stream>>>(agg1, n * 16);
  scatter_edges<<<nb(E * 16, B), B, 0, stream>>>(h1, dinv, src, dst, agg1, E, 4);
  finalize_relu<<<nb(n * 16, B), B, 0, stream>>>(h1, dinv, b1, agg1, n, 4);   // agg1 = h1relu

  // --- layer 2: h2 = h1relu @ W2 (K=16, N=8) ---
  gemm_wmma_f32<16, 8><<<nb(gthreads, B), B, 0, stream>>>(agg1, W2, h2, (int)n);
  zero_f32<<<nb(n * 8, B), B, 0, stream>>>(agg2, n * 8);
  scatter_edges<<<nb(E * 8, B), B, 0, stream>>>(h2, dinv, src, dst, agg2, E, 3);

  // --- head: relu + dot(Wd) + bd ---
  head_kernel<<<nb(n, B), B, 0, stream>>>(h2, agg2, dinv, b2, Wd, bd, out, n);
}